// Attention_71030169141419
// MI455X (gfx1250) — compile-verified
//
#include <hip/hip_runtime.h>
#include <hip/hip_bf16.h>

typedef __attribute__((ext_vector_type(2))) float v2f;
typedef __attribute__((ext_vector_type(4))) float v4f;
typedef __attribute__((ext_vector_type(8))) float v8f;

#define NHID 256
#define DHID 16
#define WAVES_PER_BLOCK 8

// gfx1250 has a hardware tanh transcendental (v_tanh_f32); use it if exposed.
static __device__ __forceinline__ float fast_tanh(float x) {
#if defined(__has_builtin)
#if __has_builtin(__builtin_amdgcn_tanhf)
    return __builtin_amdgcn_tanhf(x);
#else
    return tanhf(x);
#endif
#else
    return tanhf(x);
#endif
}

// One wave32 handles one 16-row tile.
//  - W_t/W_f/W_c staged in LDS (swizzled so each lane's B-fragment is one float2)
//  - 64 x V_WMMA_F32_16X16X4_F32 per view accumulate h = z @ W  (fp32 matrix pipe)
//  - v_tanh_f32 + q-dot via 16-lane shuffle reduction, 3-way softmax in registers
//  - weighted-sum output pass re-reads the (WGP$-hot) tile with float4 coalesced IO
__global__ __launch_bounds__(256) void attn_wmma_kernel(
    const float* __restrict__ z_t, const float* __restrict__ z_f, const float* __restrict__ z_c,
    const float* __restrict__ W_t, const float* __restrict__ b_t,
    const float* __restrict__ W_f, const float* __restrict__ b_f,
    const float* __restrict__ W_c, const float* __restrict__ b_c,
    const float* __restrict__ q,
    float* __restrict__ zOut, float* __restrict__ aOut,
    int nTiles)
{
    __shared__ float ldsW[3 * NHID * DHID];   // 48 KB, swizzled B-fragment layout

    const int tid = threadIdx.x;

    // Stage weights into LDS.
    // Swizzle: element (k, n) of view v lives at  v*4096 + (k>>1)*32 + n*2 + (k&1)
    // so the B-fragment {W[kk][n], W[kk+1][n]} for even kk is one contiguous float2.
    {
        const float* Ws[3] = {W_t, W_f, W_c};
        for (int idx = tid; idx < 3 * NHID * DHID; idx += 256) {
            int v   = idx >> 12;      // / 4096
            int rem = idx & 4095;
            int k   = rem >> 4;       // / 16
            int n   = rem & 15;
            ldsW[v * 4096 + (k >> 1) * 32 + n * 2 + (k & 1)] = Ws[v][rem];
        }
    }
    __syncthreads();

    const int wave = tid >> 5;
    const int lane = tid & 31;
    const int half = lane >> 4;   // 0: lanes 0-15, 1: lanes 16-31
    const int n    = lane & 15;
    const int tile = blockIdx.x * WAVES_PER_BLOCK + wave;
    if (tile >= nTiles) return;   // wave-uniform: EXEC stays all-ones for WMMA

    const size_t rowA = (size_t)tile * 16 + (size_t)n;   // A-matrix row for this lane
    const float* zs[3] = {z_t, z_f, z_c};

    // ---- 3 views: acc[v] = z_v_tile(16x256) @ W_v(256x16), via 16x16x4 fp32 WMMA ----
    v8f acc[3];
#pragma unroll
    for (int v = 0; v < 3; ++v) {
        v8f c = {};
        // A fragment: lane holds z[row][k + 2*half + {0,1}]
        const float* zp = zs[v] + rowA * NHID + 2 * half;
        // B fragment: lane holds {W[k+2*half][n], W[k+2*half+1][n]} from swizzled LDS
        const float* wp = ldsW + v * 4096 + half * 32 + n * 2;
#pragma unroll 4
        for (int k = 0; k < NHID; k += 4) {
            v2f a = *(const v2f*)(zp + k);
            v2f b = *(const v2f*)(wp + k * 16);   // (k>>1)*32
            c = __builtin_amdgcn_wmma_f32_16x16x4_f32(
                    false, a, false, b, (short)0, c, false, false);
        }
        acc[v] = c;
    }

    // ---- epilogue: e_v = sum_n tanh(acc + b)*q, softmax, scale & store ----
    const float bn[3] = {b_t[n], b_f[n], b_c[n]};
    const float qn    = q[n];

#pragma unroll
    for (int r = 0; r < 8; ++r) {
        // C/D layout: VGPR r holds row M=r (lanes 0-15) and M=8+r (lanes 16-31)
        float e[3];
#pragma unroll
        for (int v = 0; v < 3; ++v) {
            float h = fast_tanh(acc[v][r] + bn[v]) * qn;
            h += __shfl_xor(h, 1, 32);
            h += __shfl_xor(h, 2, 32);
            h += __shfl_xor(h, 4, 32);
            h += __shfl_xor(h, 8, 32);   // masks < 16: reduction stays per 16-lane half
            e[v] = h;
        }
        float m  = fmaxf(e[0], fmaxf(e[1], e[2]));
        float a0 = __expf(e[0] - m);
        float a1 = __expf(e[1] - m);
        float a2 = __expf(e[2] - m);
        float inv = 1.0f / (a0 + a1 + a2);
        a0 *= inv; a1 *= inv; a2 *= inv;

        const size_t row = (size_t)tile * 16 + (size_t)(r + half * 8);

        if (n < 3) {
            float av = (n == 0) ? a0 : (n == 1) ? a1 : a2;
            aOut[row * 3 + n] = av;
        }

        const float* pt = z_t + row * NHID;
        const float* pf = z_f + row * NHID;
        const float* pc = z_c + row * NHID;
        float*       po = zOut + row * NHID;
#pragma unroll
        for (int j = 0; j < 4; ++j) {
            int col = n * 4 + j * 64;  // 16 lanes/half x float4 = 256B contiguous per half
            v4f vt = *(const v4f*)(pt + col);
            v4f vf = *(const v4f*)(pf + col);
            v4f vc = *(const v4f*)(pc + col);
            v4f vo = vt * a0 + vf * a1 + vc * a2;
            *(v4f*)(po + col) = vo;
        }
    }
}

extern "C" void kernel_launch(void* const* d_in, const int* in_sizes, int n_in,
                              void* d_out, int out_size, void* d_ws, size_t ws_size,
                              hipStream_t stream) {
    const float* z_t = (const float*)d_in[0];
    const float* z_f = (const float*)d_in[1];
    const float* z_c = (const float*)d_in[2];
    const float* W_t = (const float*)d_in[3];
    const float* b_t = (const float*)d_in[4];
    const float* W_f = (const float*)d_in[5];
    const float* b_f = (const float*)d_in[6];
    const float* W_c = (const float*)d_in[7];
    const float* b_c = (const float*)d_in[8];
    const float* q   = (const float*)d_in[9];

    const int N      = in_sizes[0] / NHID;   // 200000
    const int nTiles = N / 16;               // 12500 (N divisible by 16)

    float* zOut = (float*)d_out;
    float* aOut = zOut + (size_t)N * NHID;

    const int blocks = (nTiles + WAVES_PER_BLOCK - 1) / WAVES_PER_BLOCK;
    attn_wmma_kernel<<<blocks, 256, 0, stream>>>(
        z_t, z_f, z_c, W_t, b_t, W_f, b_f, W_c, b_c, q, zOut, aOut, nTiles);
}